// Network_6339371728981
// MI455X (gfx1250) — compile-verified
//
#include <hip/hip_runtime.h>
#include <hip/hip_bf16.h>
#include <stdint.h>

#define THREADS 256
#define DT_STEP 0.02f

// ---------------------------------------------------------------------------
// One-time: pack edge list into 16B records {src, tgt, weight, pad} so the
// per-step streaming is a single b128 async load per edge.
// weight = sign * relu(syn_count) * relu(syn_strength)
// ---------------------------------------------------------------------------
__global__ void pack_edges_kernel(const float* __restrict__ sign,
                                  const float* __restrict__ syn_count,
                                  const float* __restrict__ syn_strength,
                                  const int* __restrict__ src_idx,
                                  const int* __restrict__ tgt_idx,
                                  uint4* __restrict__ pack, int E) {
    int e = blockIdx.x * blockDim.x + threadIdx.x;
    if (e >= E) return;
    float w = sign[e] * fmaxf(syn_count[e], 0.0f) * fmaxf(syn_strength[e], 0.0f);
    uint4 p;
    p.x = (unsigned)src_idx[e];
    p.y = (unsigned)tgt_idx[e];
    p.z = __float_as_uint(w);
    p.w = 0u;
    pack[e] = p;
}

// ---------------------------------------------------------------------------
// One-time: init state. Batch (B=4) is innermost -> float4 per node so the
// edge gather is one 16B load and the scatter hits one 16B segment.
// ---------------------------------------------------------------------------
__global__ void init_state_kernel(const float* __restrict__ bias,
                                  const float* __restrict__ time_const,
                                  float* __restrict__ alpha,
                                  float4* __restrict__ v4,
                                  float4* __restrict__ rates4,
                                  float4* __restrict__ acc4, int N) {
    int n = blockIdx.x * blockDim.x + threadIdx.x;
    if (n >= N) return;
    float b = bias[n];
    alpha[n] = DT_STEP / fmaxf(time_const[n], DT_STEP);
    float r = fmaxf(b, 0.0f);
    v4[n]     = make_float4(b, b, b, b);
    rates4[n] = make_float4(r, r, r, r);
    acc4[n]   = make_float4(0.f, 0.f, 0.f, 0.f);
}

// ---------------------------------------------------------------------------
// Per-step edge kernel: grid-stride over 256-edge tiles. Edge records are
// streamed L2 -> LDS with double-buffered global_load_async_to_lds_b128
// (CDNA5 async path, ASYNCcnt), consumed from LDS, then one 16B gather of
// rates and 4 native f32 atomics (device scope) into the accumulator.
// ---------------------------------------------------------------------------
__global__ __launch_bounds__(THREADS) void edge_kernel(
        const uint4* __restrict__ pack,
        const float4* __restrict__ rates4,
        float4* __restrict__ acc4,
        int E, int ntiles) {
    __shared__ uint4 stage[2][THREADS];
    const int tid = threadIdx.x;
    const int stride = gridDim.x;

    auto issue = [&](int tile, int buf) {
        int e = tile * THREADS + tid;
        if (e < E) {
            unsigned lds = (unsigned)(uintptr_t)(&stage[buf][tid]);
            unsigned go  = (unsigned)e * 16u;
            // vdst = LDS byte address, vaddr = 32-bit per-lane offset (GVS),
            // saddr = 64-bit base SGPR pair
            asm volatile("global_load_async_to_lds_b128 %0, %1, %2"
                         :: "v"(lds), "v"(go), "s"(pack) : "memory");
        }
    };

    int tile = blockIdx.x;
    int buf = 0;
    if (tile < ntiles) issue(tile, 0);
    for (; tile < ntiles; tile += stride) {
        int nxt = tile + stride;
        if (nxt < ntiles) {
            issue(nxt, buf ^ 1);                              // 2 outstanding
            asm volatile("s_wait_asynccnt 0x1" ::: "memory"); // current done
        } else {
            asm volatile("s_wait_asynccnt 0x0" ::: "memory");
        }
        int e = tile * THREADS + tid;
        if (e < E) {
            uint4 d = stage[buf][tid];
            int   s = (int)d.x;
            float w = __uint_as_float(d.z);
            float4 r = rates4[s];                             // 16B L2 gather
            unsigned boff = d.y * 16u;                        // acc4[tgt] byte offset
            float m0 = r.x * w, m1 = r.y * w, m2 = r.z * w, m3 = r.w * w;
            // Native no-return f32 atomics into L2 at device scope.
            asm volatile("global_atomic_add_f32 %0, %1, %2 scope:SCOPE_DEV"
                         :: "v"(boff), "v"(m0), "s"(acc4) : "memory");
            asm volatile("global_atomic_add_f32 %0, %1, %2 offset:4 scope:SCOPE_DEV"
                         :: "v"(boff), "v"(m1), "s"(acc4) : "memory");
            asm volatile("global_atomic_add_f32 %0, %1, %2 offset:8 scope:SCOPE_DEV"
                         :: "v"(boff), "v"(m2), "s"(acc4) : "memory");
            asm volatile("global_atomic_add_f32 %0, %1, %2 offset:12 scope:SCOPE_DEV"
                         :: "v"(boff), "v"(m3), "s"(acc4) : "memory");
        }
        buf ^= 1;
    }
}

// ---------------------------------------------------------------------------
// Per-step Euler update: v += alpha*(-v + bias + acc + x_t); emit relu(v) to
// out[B,T,N], refresh rates, and zero the accumulator for the next step.
// ---------------------------------------------------------------------------
__global__ __launch_bounds__(THREADS) void update_kernel(
        const float* __restrict__ x,      // [B,T,N]
        const float* __restrict__ bias,
        const float* __restrict__ alpha,
        float4* __restrict__ v4,
        float4* __restrict__ rates4,
        float4* __restrict__ acc4,
        float* __restrict__ out,          // [B,T,N]
        int N, int T, int t) {
    int n = blockIdx.x * blockDim.x + threadIdx.x;
    if (n >= N) return;
    float  a  = alpha[n];
    float  b  = bias[n];
    float4 v  = v4[n];
    float4 s  = acc4[n];
    size_t TN   = (size_t)T * (size_t)N;
    size_t base = (size_t)t * (size_t)N + (size_t)n;
    float x0 = x[base];
    float x1 = x[base + TN];
    float x2 = x[base + 2 * TN];
    float x3 = x[base + 3 * TN];
    v.x += a * (b + s.x + x0 - v.x);
    v.y += a * (b + s.y + x1 - v.y);
    v.z += a * (b + s.z + x2 - v.z);
    v.w += a * (b + s.w + x3 - v.w);
    float4 r = make_float4(fmaxf(v.x, 0.f), fmaxf(v.y, 0.f),
                           fmaxf(v.z, 0.f), fmaxf(v.w, 0.f));
    v4[n]     = v;
    rates4[n] = r;
    acc4[n]   = make_float4(0.f, 0.f, 0.f, 0.f);
    out[base]          = r.x;
    out[base + TN]     = r.y;
    out[base + 2 * TN] = r.z;
    out[base + 3 * TN] = r.w;
}

extern "C" void kernel_launch(void* const* d_in, const int* in_sizes, int n_in,
                              void* d_out, int out_size, void* d_ws, size_t ws_size,
                              hipStream_t stream) {
    (void)n_in; (void)out_size; (void)ws_size;
    const float* x    = (const float*)d_in[0];
    const float* bias = (const float*)d_in[1];
    const float* tc   = (const float*)d_in[2];
    const float* sign = (const float*)d_in[3];
    const float* sc   = (const float*)d_in[4];
    const float* ss   = (const float*)d_in[5];
    const int*   src  = (const int*)d_in[6];
    const int*   tgt  = (const int*)d_in[7];
    float*       out  = (float*)d_out;

    const int N = in_sizes[1];
    const int E = in_sizes[3];
    const int B = 4;
    const int T = in_sizes[0] / (B * N);

    // Workspace carve-out (256B aligned): pack 25.6MB + state ~3.2MB
    char* ws = (char*)d_ws;
    size_t off = 0;
    auto carve = [&](size_t bytes) {
        void* p = ws + off;
        off = (off + bytes + 255) & ~(size_t)255;
        return p;
    };
    uint4*  pack   = (uint4*) carve((size_t)E * sizeof(uint4));
    float*  alpha  = (float*) carve((size_t)N * sizeof(float));
    float4* v4     = (float4*)carve((size_t)N * sizeof(float4));
    float4* rates4 = (float4*)carve((size_t)N * sizeof(float4));
    float4* acc4   = (float4*)carve((size_t)N * sizeof(float4));

    const int egrid  = (E + THREADS - 1) / THREADS;  // edge tiles
    const int ngrid  = (N + THREADS - 1) / THREADS;
    pack_edges_kernel<<<egrid, THREADS, 0, stream>>>(sign, sc, ss, src, tgt, pack, E);
    init_state_kernel<<<ngrid, THREADS, 0, stream>>>(bias, tc, alpha, v4, rates4, acc4, N);

    const int ntiles  = egrid;
    const int eblocks = ntiles < 2048 ? ntiles : 2048;
    for (int t = 0; t < T; ++t) {
        edge_kernel<<<eblocks, THREADS, 0, stream>>>(pack, rates4, acc4, E, ntiles);
        update_kernel<<<ngrid, THREADS, 0, stream>>>(x, bias, alpha, v4, rates4, acc4,
                                                     out, N, T, t);
    }
}